// MMD_loss_42640435314741
// MI455X (gfx1250) — compile-verified
//
#include <hip/hip_runtime.h>
#include <hip/hip_bf16.h>

typedef __attribute__((ext_vector_type(2))) float v2f;
typedef __attribute__((ext_vector_type(8))) float v8f;

#define N_HALF 4096
#define M_TOT  8192
#define DDIM   512
#define KC     32
#define LDA    36   // padded LDS row stride (floats): 16-row frag reads hit 16 distinct banks, 16B-aligned float4 stores

// workspace layout (float indices)
#define SQ_OFF     0      // 8192 floats: per-row sum of squares
#define COL_OFF    8192   // 512 floats: column sums of T
#define SUMSQ_OFF  8704   // 1 float
#define NEGINV_OFF 8708   // 5 floats: -1/(bw*2^q)
#define SUMD_OFF   8720   // double (byte offset 34880, 8-aligned)

__device__ __forceinline__ const float* t_row(const float* __restrict__ src,
                                              const float* __restrict__ tgt, int r) {
    return (r < N_HALF) ? (src + (size_t)r * DDIM)
                        : (tgt + (size_t)(r - N_HALF) * DDIM);
}

__global__ void init_kernel(float* ws) {
    int t = threadIdx.x;
    for (int i = t; i < 512; i += 256) ws[COL_OFF + i] = 0.0f;
    if (t == 0) {
        ws[SUMSQ_OFF] = 0.0f;
        ws[SUMD_OFF] = 0.0f;
        ws[SUMD_OFF + 1] = 0.0f;  // double accumulator = 0
    }
}

// One wave per row: sq[r] = sum(T[r]^2); also accumulate total sum of sq.
__global__ void rowsq_kernel(const float* __restrict__ src, const float* __restrict__ tgt,
                             float* ws) {
    int wave = threadIdx.x >> 5, lane = threadIdx.x & 31;
    int r = blockIdx.x * 8 + wave;
    const float* row = t_row(src, tgt, r);
    float acc = 0.0f;
#pragma unroll
    for (int t = 0; t < 4; ++t) {
        float4 v = ((const float4*)row)[lane + 32 * t];
        acc += v.x * v.x + v.y * v.y + v.z * v.z + v.w * v.w;
    }
    for (int off = 16; off; off >>= 1) acc += __shfl_xor(acc, off, 32);
    if (lane == 0) {
        ws[SQ_OFF + r] = acc;
        atomicAdd(ws + SUMSQ_OFF, acc);
    }
}

// Column sums of T (for ||colsum||^2 -> analytic sum(L2), no GEMM needed).
__global__ void colsum_kernel(const float* __restrict__ src, const float* __restrict__ tgt,
                              float* ws) {
    int t = threadIdx.x;
    int r0 = blockIdx.x * 128;
    float a0 = 0.0f, a1 = 0.0f;
    for (int rr = 0; rr < 128; ++rr) {
        const float* row = t_row(src, tgt, r0 + rr);
        a0 += row[t];
        a1 += row[t + 256];
    }
    atomicAdd(ws + COL_OFF + t, a0);
    atomicAdd(ws + COL_OFF + t + 256, a1);
}

// sum(L2) = 2*m*sum(sq) - 2*||colsum||^2 ; bandwidth = sum(L2)/(m^2-m)/KERNEL_MUL^2
__global__ void bw_kernel(float* ws) {
    __shared__ float red[8];
    int t = threadIdx.x;
    float s0 = ws[COL_OFF + t], s1 = ws[COL_OFF + t + 256];
    float acc = s0 * s0 + s1 * s1;
    for (int off = 16; off; off >>= 1) acc += __shfl_xor(acc, off, 32);
    if ((t & 31) == 0) red[t >> 5] = acc;
    __syncthreads();
    if (t == 0) {
        float s2 = 0.0f;
#pragma unroll
        for (int i = 0; i < 8; ++i) s2 += red[i];
        double m = (double)M_TOT;
        double sumL2 = 2.0 * m * (double)ws[SUMSQ_OFF] - 2.0 * (double)s2;
        double bw = sumL2 / (m * m - m) / 4.0;  // KERNEL_MUL^(KERNEL_NUM//2) = 2^2
#pragma unroll
        for (int q = 0; q < 5; ++q)
            ws[NEGINV_OFF + q] = (float)(-1.0 / (bw * (double)(1 << q)));
    }
}

// Fused GEMM (f32 WMMA) + L2 + 5-Gaussian sum + signed reduction.
// Grid 64x64 blocks of 128x128 output; only bj>=bi computed (symmetry, weight 2).
// Staging is direct global->LDS (no register prefetch: avoids scratch spills;
// ~8 blocks/WGP of occupancy hides the staging latency behind other blocks' WMMAs).
__global__ void __launch_bounds__(256) mmd_kernel(const float* __restrict__ src,
                                                  const float* __restrict__ tgt,
                                                  float* ws) {
    int bi = blockIdx.x >> 6, bj = blockIdx.x & 63;
    if (bj < bi) return;

    __shared__ float As[128 * LDA];
    __shared__ float Bs[128 * LDA];
    __shared__ float sqi[128], sqj[128];
    __shared__ float red[8];

    const int tid = threadIdx.x;
    const int wave = tid >> 5, lane = tid & 31;
    const int i0 = bi * 128, j0 = bj * 128;

    if (tid < 128) sqi[tid] = ws[SQ_OFF + i0 + tid];
    else           sqj[tid - 128] = ws[SQ_OFF + j0 + tid - 128];

    const int rp = wave & 3;   // tile-row pair: rows rp*32, rp*32+16
    const int cg = wave >> 2;  // tile-col quad: cols cg*64 + 16*tj
    const int fr = lane & 15, hi = lane >> 4;

    v8f c[2][4];
#pragma unroll
    for (int a = 0; a < 2; ++a)
#pragma unroll
        for (int b = 0; b < 4; ++b)
#pragma unroll
            for (int e = 0; e < 8; ++e) c[a][b][e] = 0.0f;

    for (int kc = 0; kc < DDIM; kc += KC) {
        __syncthreads();
#pragma unroll
        for (int l = 0; l < 4; ++l) {
            int idx = tid + 256 * l;
            int rr = idx >> 3, c4 = (idx & 7) * 4;
            const float* ra = t_row(src, tgt, i0 + rr);
            *(float4*)&As[rr * LDA + c4] = *(const float4*)(ra + kc + c4);
            const float* rb = t_row(src, tgt, j0 + rr);
            *(float4*)&Bs[rr * LDA + c4] = *(const float4*)(rb + kc + c4);
        }
        __syncthreads();
#pragma unroll
        for (int kk = 0; kk < KC; kk += 4) {
            // A 16x4 f32 frag: lane(0-15)->M=lane,K=vgpr; lane(16-31)->K=vgpr+2.
            // B 4x16 f32 frag has the identical lane->(row-of-T, K) map for C = T*T^T.
            int col = kk + 2 * hi;
            v2f a[2], b[4];
#pragma unroll
            for (int ti = 0; ti < 2; ++ti)
                a[ti] = *(const v2f*)&As[(rp * 32 + ti * 16 + fr) * LDA + col];
#pragma unroll
            for (int tj = 0; tj < 4; ++tj)
                b[tj] = *(const v2f*)&Bs[(cg * 64 + tj * 16 + fr) * LDA + col];
#pragma unroll
            for (int ti = 0; ti < 2; ++ti)
#pragma unroll
                for (int tj = 0; tj < 4; ++tj)
                    c[ti][tj] = __builtin_amdgcn_wmma_f32_16x16x4_f32(
                        false, a[ti], false, b[tj], (short)0, c[ti][tj], false, false);
        }
    }

    float ninv[5];
#pragma unroll
    for (int q = 0; q < 5; ++q) ninv[q] = ws[NEGINV_OFF + q];

    float acc = 0.0f;
#pragma unroll
    for (int ti = 0; ti < 2; ++ti) {
#pragma unroll
        for (int tj = 0; tj < 4; ++tj) {
            float sj = sqj[cg * 64 + tj * 16 + fr];   // N = lane&15
#pragma unroll
            for (int r = 0; r < 8; ++r) {
                float si = sqi[rp * 32 + ti * 16 + r + 8 * hi];  // M = r + 8*(lane>>4)
                float l2 = si + sj - 2.0f * c[ti][tj][r];
#pragma unroll
                for (int q = 0; q < 5; ++q) acc += __expf(l2 * ninv[q]);
            }
        }
    }

    float sgn = ((i0 < N_HALF) == (j0 < N_HALF)) ? 1.0f : -1.0f;
    float wgt = (bi == bj) ? 1.0f : 2.0f;
    for (int off = 16; off; off >>= 1) acc += __shfl_xor(acc, off, 32);
    if (lane == 0) red[wave] = acc;
    __syncthreads();
    if (tid == 0) {
        float tot = 0.0f;
#pragma unroll
        for (int i = 0; i < 8; ++i) tot += red[i];
        atomicAdd((double*)(ws + SUMD_OFF), (double)(sgn * wgt) * (double)tot);
    }
}

__global__ void finalize_kernel(const float* ws, float* out) {
    double s = *(const double*)(ws + SUMD_OFF);
    out[0] = (float)(s / ((double)N_HALF * (double)N_HALF));
}

extern "C" void kernel_launch(void* const* d_in, const int* in_sizes, int n_in,
                              void* d_out, int out_size, void* d_ws, size_t ws_size,
                              hipStream_t stream) {
    const float* src = (const float*)d_in[0];
    const float* tgt = (const float*)d_in[1];
    float* ws = (float*)d_ws;
    float* out = (float*)d_out;

    init_kernel<<<1, 256, 0, stream>>>(ws);
    rowsq_kernel<<<M_TOT / 8, 256, 0, stream>>>(src, tgt, ws);
    colsum_kernel<<<M_TOT / 128, 256, 0, stream>>>(src, tgt, ws);
    bw_kernel<<<1, 256, 0, stream>>>(ws);
    mmd_kernel<<<64 * 64, 256, 0, stream>>>(src, tgt, ws);
    finalize_kernel<<<1, 1, 0, stream>>>(ws, out);
}